// DAEGC_19791209300766
// MI455X (gfx1250) — compile-verified
//
#include <hip/hip_runtime.h>
#include <hip/hip_bf16.h>
#include <math.h>

// ---------------------------------------------------------------------------
// DAEGC forward on MI455X (gfx1250): fp32 WMMA (V_WMMA_F32_16X16X4_F32) for
// every GEMM; flash-attention-style fused GAT attention (single pass over the
// 256MB M / adj matrices per layer) with fully distributed online softmax.
// Shapes fixed by reference: N=8192, F=512, H=256, E=64, K=16.
// ---------------------------------------------------------------------------

#define N_NODES 8192
#define F_IN    512
#define H_DIM   256
#define E_DIM   64
#define K_CLUS  16
#define NEG_BIG -9.0e15f

typedef float v2f __attribute__((ext_vector_type(2)));
typedef float v8f __attribute__((ext_vector_type(8)));

__device__ __forceinline__ v8f wmma_f32(v2f a, v2f b, v8f c) {
    // 8 args: (neg_a, A, neg_b, B, c_mod, C, reuse_a, reuse_b)
    return __builtin_amdgcn_wmma_f32_16x16x4_f32(false, a, false, b,
                                                 (short)0, c, false, false);
}

// ---------------------------------------------------------------------------
// Dense GEMM  C[M x Ncols] = A[M x Kdim] @ B[Kdim x Ncols], fp32 WMMA.
// Block = 128 threads (4 waves); block computes 16 rows x (4*CT*16) cols.
// ---------------------------------------------------------------------------
template <int CT>
__global__ void gemm_wmma_f32(const float* __restrict__ A,
                              const float* __restrict__ B,
                              float* __restrict__ C,
                              int Kdim, int Ncols) {
    const int lane = threadIdx.x & 31;
    const int wave = threadIdx.x >> 5;
    const int half = lane >> 4;       // 0: K low pair, 1: K high pair
    const int l16  = lane & 15;
    const int rbase = blockIdx.x * 16;

    v8f acc[CT];
    #pragma unroll
    for (int t = 0; t < CT; ++t)
        #pragma unroll
        for (int g = 0; g < 8; ++g) acc[t][g] = 0.0f;

    int cols[CT];
    #pragma unroll
    for (int t = 0; t < CT; ++t) cols[t] = (wave * CT + t) * 16 + l16;

    const float* arow = A + (size_t)(rbase + l16) * Kdim;

    for (int k = 0; k < Kdim; k += 4) {
        const int ka = k + 2 * half;
        v2f a;                         // A 16x4 f32 fragment (ISA layout)
        a.x = arow[ka];
        a.y = arow[ka + 1];
        #pragma unroll
        for (int t = 0; t < CT; ++t) {
            v2f b;                     // B 4x16 f32 fragment
            b.x = B[(size_t)ka * Ncols + cols[t]];
            b.y = B[(size_t)(ka + 1) * Ncols + cols[t]];
            acc[t] = wmma_f32(a, b, acc[t]);
        }
    }

    #pragma unroll
    for (int t = 0; t < CT; ++t)
        #pragma unroll
        for (int g = 0; g < 8; ++g)
            C[(size_t)(rbase + g + 8 * half) * Ncols + cols[t]] = acc[t][g];
}

// ---------------------------------------------------------------------------
// Per-node attention coefficients: as[i] = h[i]·a_self, an[i] = h[i]·a_neigh.
// One wave per row.
// ---------------------------------------------------------------------------
__global__ void attn_coef(const float* __restrict__ h,
                          const float* __restrict__ a_s,
                          const float* __restrict__ a_n,
                          float* __restrict__ outs,
                          float* __restrict__ outn, int Hd) {
    const int lane = threadIdx.x & 31;
    const int wave = threadIdx.x >> 5;
    const int row  = blockIdx.x * 4 + wave;
    const float* hr = h + (size_t)row * Hd;
    float ss = 0.0f, sn = 0.0f;
    for (int e = lane; e < Hd; e += 32) {
        const float v = hr[e];
        ss += v * a_s[e];
        sn += v * a_n[e];
    }
    #pragma unroll
    for (int m = 16; m >= 1; m >>= 1) {
        ss += __shfl_xor(ss, m, 32);
        sn += __shfl_xor(sn, m, 32);
    }
    if (lane == 0) { outs[row] = ss; outn[row] = sn; }
}

// ---------------------------------------------------------------------------
// Fused GAT attention: for a 16-row stripe, stream 64-column chunks of the
// 8192-wide attention row:  score -> LeakyReLU -> adj mask -> online softmax
// -> WMMA accumulate P @ h -> ELU.  M/adj read exactly once.
//
// Block = 128 threads (4 waves). Wave w owns output columns [w*HD/4,...).
// Score producers: thread t -> row t>>3, cols (t&7)*8..+7; the 8 threads of
// a row are contiguous lanes, so row max/sum reduce with 3 shfl_xor steps and
// the running softmax state (m_run, denom) is kept redundantly (identical)
// in all 8 threads of the row.
// ---------------------------------------------------------------------------
template <int HD>
__global__ void gat_attn(const float* __restrict__ h,
                         const float* __restrict__ as_,
                         const float* __restrict__ an_,
                         const float* __restrict__ Mm,
                         const int* __restrict__ adj,
                         float* __restrict__ out) {
    constexpr int TPW = HD / 64;          // 16x16 C tiles per wave
    __shared__ float p[16][68];           // P tile, padded stride (no bank conflicts)
    __shared__ float sc_lds[16];          // per-row rescale factors
    __shared__ float dn_lds[16];          // per-row softmax denominators

    const int t    = threadIdx.x;
    const int lane = t & 31;
    const int wave = t >> 5;
    const int half = lane >> 4;
    const int l16  = lane & 15;
    const int rbase = blockIdx.x * 16;

    // score-producer mapping
    const int srow = t >> 3;
    const int scg  = (t & 7) * 8;
    const float as_r = as_[rbase + srow];
    const float* Mrow = Mm  + (size_t)(rbase + srow) * N_NODES;
    const int*   Arow = adj + (size_t)(rbase + srow) * N_NODES;

    v8f acc[TPW];
    #pragma unroll
    for (int tt = 0; tt < TPW; ++tt)
        #pragma unroll
        for (int g = 0; g < 8; ++g) acc[tt][g] = 0.0f;

    int cols[TPW];
    #pragma unroll
    for (int tt = 0; tt < TPW; ++tt) cols[tt] = (wave * TPW + tt) * 16 + l16;

    float m_run = -3.0e38f;   // running row max (redundant per 8-thread row group)
    float denom = 0.0f;       // running softmax denominator (redundant)

    for (int cb = 0; cb < N_NODES; cb += 64) {
        // ---- 1. scores in registers -------------------------------------
        const float4 m0 = *(const float4*)(Mrow + cb + scg);
        const float4 m1 = *(const float4*)(Mrow + cb + scg + 4);
        const int4   a0 = *(const int4*)(Arow + cb + scg);
        const int4   a1 = *(const int4*)(Arow + cb + scg + 4);
        if (cb + 64 < N_NODES) {           // CDNA5 prefetch of next chunk
            __builtin_prefetch(Mrow + cb + 64 + scg, 0, 0);
            __builtin_prefetch(Arow + cb + 64 + scg, 0, 0);
        }
        const float mv[8] = {m0.x, m0.y, m0.z, m0.w, m1.x, m1.y, m1.z, m1.w};
        const int   av[8] = {a0.x, a0.y, a0.z, a0.w, a1.x, a1.y, a1.z, a1.w};
        float sv[8];
        #pragma unroll
        for (int j = 0; j < 8; ++j) {
            float s = (as_r + an_[cb + scg + j]) * mv[j];
            s = (s > 0.0f) ? s : 0.2f * s;          // LeakyReLU(0.2)
            sv[j] = (av[j] > 0) ? s : NEG_BIG;      // adjacency mask
        }

        // ---- 2. distributed online softmax ------------------------------
        float tmax = sv[0];
        #pragma unroll
        for (int j = 1; j < 8; ++j) tmax = fmaxf(tmax, sv[j]);
        #pragma unroll
        for (int m = 4; m >= 1; m >>= 1)            // reduce across 8-lane row group
            tmax = fmaxf(tmax, __shfl_xor(tmax, m, 32));

        const float nm    = fmaxf(m_run, tmax);
        const float scale = __expf(m_run - nm);
        float lsum = 0.0f;
        #pragma unroll
        for (int j = 0; j < 8; ++j) {
            const float pv = __expf(sv[j] - nm);
            p[srow][scg + j] = pv;
            lsum += pv;
        }
        #pragma unroll
        for (int m = 4; m >= 1; m >>= 1)
            lsum += __shfl_xor(lsum, m, 32);

        denom = denom * scale + lsum;
        m_run = nm;
        if ((t & 7) == 0) sc_lds[srow] = scale;
        __syncthreads();

        // ---- 3. rescale C fragments (rows g + 8*half) -------------------
        float s8[8];
        #pragma unroll
        for (int g = 0; g < 8; ++g) s8[g] = sc_lds[g + 8 * half];
        #pragma unroll
        for (int tt = 0; tt < TPW; ++tt)
            #pragma unroll
            for (int g = 0; g < 8; ++g) acc[tt][g] *= s8[g];

        // ---- 4. WMMA: acc += P(16x64) @ h(64xHD slice) ------------------
        #pragma unroll
        for (int kk = 0; kk < 16; ++kk) {
            const int ko = kk * 4 + 2 * half;
            v2f a;                              // A fragment from LDS P tile
            a.x = p[l16][ko];
            a.y = p[l16][ko + 1];
            #pragma unroll
            for (int tt = 0; tt < TPW; ++tt) {
                v2f b;                          // B fragment streamed from h
                b.x = h[(size_t)(cb + ko) * HD + cols[tt]];
                b.y = h[(size_t)(cb + ko + 1) * HD + cols[tt]];
                acc[tt] = wmma_f32(a, b, acc[tt]);
            }
        }
        __syncthreads();   // protect P tile before next chunk overwrites it
    }

    if ((t & 7) == 0) dn_lds[srow] = denom;
    __syncthreads();

    float d8[8];
    #pragma unroll
    for (int g = 0; g < 8; ++g) d8[g] = dn_lds[g + 8 * half];

    #pragma unroll
    for (int tt = 0; tt < TPW; ++tt)
        #pragma unroll
        for (int g = 0; g < 8; ++g) {
            float v = acc[tt][g] / d8[g];
            v = (v > 0.0f) ? v : (__expf(v) - 1.0f);    // ELU (alpha=1)
            out[(size_t)(rbase + g + 8 * half) * HD + cols[tt]] = v;
        }
}

// ---------------------------------------------------------------------------
// Row L2-normalize (E=64): z = g / max(||g||2, 1e-12). One wave per row.
// ---------------------------------------------------------------------------
__global__ void l2norm_rows(const float* __restrict__ g, float* __restrict__ z) {
    const int lane = threadIdx.x & 31;
    const int wave = threadIdx.x >> 5;
    const int row  = blockIdx.x * 4 + wave;
    const float* gr = g + (size_t)row * E_DIM;
    const float v0 = gr[lane];
    const float v1 = gr[lane + 32];
    float ss = v0 * v0 + v1 * v1;
    #pragma unroll
    for (int m = 16; m >= 1; m >>= 1) ss += __shfl_xor(ss, m, 32);
    const float nrm = fmaxf(sqrtf(ss), 1e-12f);
    z[(size_t)row * E_DIM + lane]      = v0 / nrm;
    z[(size_t)row * E_DIM + lane + 32] = v1 / nrm;
}

// ---------------------------------------------------------------------------
// A_pred = sigmoid(z @ z^T), K=64, fp32 WMMA. Block: 16 rows x 64 cols.
// ---------------------------------------------------------------------------
__global__ void apred_sigmoid(const float* __restrict__ z, float* __restrict__ Ap) {
    const int lane = threadIdx.x & 31;
    const int wave = threadIdx.x >> 5;
    const int half = lane >> 4;
    const int l16  = lane & 15;
    const int rbase = blockIdx.x * 16;
    const int col   = blockIdx.y * 64 + wave * 16 + l16;

    const float* arow = z + (size_t)(rbase + l16) * E_DIM;
    const float* brow = z + (size_t)col * E_DIM;     // B = z^T

    v8f acc;
    #pragma unroll
    for (int g = 0; g < 8; ++g) acc[g] = 0.0f;

    #pragma unroll
    for (int k = 0; k < E_DIM; k += 4) {
        const int ka = k + 2 * half;
        v2f a; a.x = arow[ka]; a.y = arow[ka + 1];
        v2f b; b.x = brow[ka]; b.y = brow[ka + 1];
        acc = wmma_f32(a, b, acc);
    }

    #pragma unroll
    for (int g = 0; g < 8; ++g) {
        const float v = acc[g];
        Ap[(size_t)(rbase + g + 8 * half) * N_NODES + col] =
            1.0f / (1.0f + __expf(-v));
    }
}

// ---------------------------------------------------------------------------
// Student-t cluster assignment, V=1: q = (1/(1+d2)) row-normalized.
// One thread per node row.
// ---------------------------------------------------------------------------
__global__ void student_t_q(const float* __restrict__ z,
                            const float* __restrict__ cc,
                            float* __restrict__ q) {
    const int row = blockIdx.x * blockDim.x + threadIdx.x;
    float zr[E_DIM];
    #pragma unroll
    for (int e = 0; e < E_DIM; ++e) zr[e] = z[(size_t)row * E_DIM + e];
    float qs[K_CLUS];
    float ssum = 0.0f;
    #pragma unroll
    for (int k = 0; k < K_CLUS; ++k) {
        float d2 = 0.0f;
        #pragma unroll
        for (int e = 0; e < E_DIM; ++e) {
            const float d = zr[e] - cc[k * E_DIM + e];
            d2 += d * d;
        }
        const float qq = 1.0f / (1.0f + d2);   // (1/(1+d2/V))^((V+1)/2), V=1
        qs[k] = qq;
        ssum += qq;
    }
    #pragma unroll
    for (int k = 0; k < K_CLUS; ++k) q[(size_t)row * K_CLUS + k] = qs[k] / ssum;
}

// ---------------------------------------------------------------------------
extern "C" void kernel_launch(void* const* d_in, const int* in_sizes, int n_in,
                              void* d_out, int out_size, void* d_ws, size_t ws_size,
                              hipStream_t stream) {
    const float* x    = (const float*)d_in[0];   // [N, F]
    const int*   adj  = (const int*)  d_in[1];   // [N, N]
    const float* Mm   = (const float*)d_in[2];   // [N, N]
    const float* W1   = (const float*)d_in[3];   // [F, H]
    const float* as1v = (const float*)d_in[4];   // [H]
    const float* an1v = (const float*)d_in[5];   // [H]
    const float* W2   = (const float*)d_in[6];   // [H, E]
    const float* as2v = (const float*)d_in[7];   // [E]
    const float* an2v = (const float*)d_in[8];   // [E]
    const float* cc   = (const float*)d_in[9];   // [K, E]

    // workspace layout (floats)
    float* h1  = (float*)d_ws;                                // N*H
    float* g1  = h1  + (size_t)N_NODES * H_DIM;               // N*H
    float* h2  = g1  + (size_t)N_NODES * H_DIM;               // N*E
    float* g2  = h2  + (size_t)N_NODES * E_DIM;               // N*E
    float* as1 = g2  + (size_t)N_NODES * E_DIM;               // N
    float* an1 = as1 + N_NODES;                               // N
    float* as2 = an1 + N_NODES;                               // N
    float* an2 = as2 + N_NODES;                               // N

    // output layout: A_pred [N*N], z [N*E], q [N*K]
    float* Ap = (float*)d_out;
    float* z  = Ap + (size_t)N_NODES * N_NODES;
    float* q  = z  + (size_t)N_NODES * E_DIM;

    const dim3 blk128(128);

    // ---- Layer 1 -------------------------------------------------------
    gemm_wmma_f32<4><<<dim3(N_NODES / 16), blk128, 0, stream>>>(x, W1, h1,
                                                                F_IN, H_DIM);
    attn_coef<<<dim3(N_NODES / 4), blk128, 0, stream>>>(h1, as1v, an1v,
                                                        as1, an1, H_DIM);
    gat_attn<H_DIM><<<dim3(N_NODES / 16), blk128, 0, stream>>>(h1, as1, an1,
                                                               Mm, adj, g1);

    // ---- Layer 2 -------------------------------------------------------
    gemm_wmma_f32<1><<<dim3(N_NODES / 16), blk128, 0, stream>>>(g1, W2, h2,
                                                                H_DIM, E_DIM);
    attn_coef<<<dim3(N_NODES / 4), blk128, 0, stream>>>(h2, as2v, an2v,
                                                        as2, an2, E_DIM);
    gat_attn<E_DIM><<<dim3(N_NODES / 16), blk128, 0, stream>>>(h2, as2, an2,
                                                               Mm, adj, g2);

    // ---- Heads ---------------------------------------------------------
    l2norm_rows<<<dim3(N_NODES / 4), blk128, 0, stream>>>(g2, z);
    apred_sigmoid<<<dim3(N_NODES / 16, N_NODES / 64), blk128, 0, stream>>>(z, Ap);
    student_t_q<<<dim3(N_NODES / 256), dim3(256), 0, stream>>>(z, cc, q);
}